// GraphNeuralNetworkSimple_29300266893460
// MI455X (gfx1250) — compile-verified
//
#include <hip/hip_runtime.h>
#include <hip/hip_bf16.h>

// ---------------------------------------------------------------------------
// GNN (3-layer edge-conv style) for MI455X / gfx1250, wave32, FP32 WMMA.
//   h = x @ proj_w + proj_b
//   repeat 3x:
//     m   = leakyrelu(concat(h[src], edge_attr) @ u2_w[l] + u2_b[l], 0.1)
//     agg = scatter_add(m, dst)
//     h   = concat(h, agg) @ u1_w[l] + u1_b[l]
// Uses V_WMMA_F32_16X16X4_F32 (exact f32, matches reference numerics).
// ---------------------------------------------------------------------------

typedef float v2f __attribute__((ext_vector_type(2)));
typedef float v8f __attribute__((ext_vector_type(8)));

#define HC   64      // node hidden
#define HB   16      // bond hidden
#define KZ   80      // HC + HB
#define K1   128     // 2*HC
#define ZSTR 84      // LDS row stride for z tile (16B aligned, bank-spread)

// ---------------------------------------------------------------------------
// Projection: h0[n,c] = sum_k x[n,k] * pw[k,c] + pb[c]     (K=75 -> VALU)
// ---------------------------------------------------------------------------
__global__ void gnn_proj_kernel(const float* __restrict__ x,
                                const float* __restrict__ pw,
                                const float* __restrict__ pb,
                                float* __restrict__ h0,
                                int nNodes, int inDim) {
    __shared__ float sPW[75 * HC];
    const int tid = threadIdx.x;
    for (int i = tid; i < inDim * HC; i += blockDim.x) sPW[i] = pw[i];
    __syncthreads();

    const long t = (long)blockIdx.x * blockDim.x + tid;
    if (t < (long)nNodes * HC) {
        const int n = (int)(t >> 6);
        const int c = (int)(t & 63);
        float acc = pb[c];
        const float* xr = x + (long)n * inDim;
        #pragma unroll 5
        for (int k = 0; k < inDim; ++k)
            acc = fmaf(xr[k], sPW[k * HC + c], acc);
        h0[t] = acc;
    }
}

// ---------------------------------------------------------------------------
// Edge message + scatter-add.  128 threads = 4 waves, each wave owns a
// 16-edge tile.  D[16x64] = leakyrelu(Z[16x80] @ W2[80x64] + b2), then
// global_atomic_add_f32 into agg[dst].
// ---------------------------------------------------------------------------
__global__ void gnn_edge_kernel(const float* __restrict__ h,
                                const int*   __restrict__ src,
                                const int*   __restrict__ dst,
                                const float* __restrict__ eattr,
                                const float* __restrict__ w2,   // [80,64]
                                const float* __restrict__ b2,   // [64]
                                float* __restrict__ agg,
                                int nEdges) {
    __shared__ float sW2[KZ * HC];        // 20 KB
    __shared__ float sZ[4][16 * ZSTR];    // 21 KB
    __shared__ int   sDst[4][16];

    const int tid  = threadIdx.x;
    const int lane = tid & 31;
    const int wave = tid >> 5;

    // Stage weights (whole block cooperates).
    for (int i = tid; i < KZ * HC; i += blockDim.x) sW2[i] = w2[i];

    // Stage this wave's 16-edge z-tile: [h[src] (64) | edge_attr (16)].
    const int tileBase = blockIdx.x * 64 + wave * 16;
    if (lane < 16) {
        const int e = tileBase + lane;
        sDst[wave][lane] = (e < nEdges) ? dst[e] : -1;
    }
    for (int i = 0; i < 16; ++i) {
        const int e = tileBase + i;
        float v0 = 0.f, v1 = 0.f;
        if (e < nEdges) {
            const long s = (long)src[e];
            v0 = h[s * HC + lane];
            v1 = h[s * HC + 32 + lane];
        }
        sZ[wave][i * ZSTR + lane]      = v0;
        sZ[wave][i * ZSTR + 32 + lane] = v1;
        if (lane < 16) {
            sZ[wave][i * ZSTR + 64 + lane] =
                (e < nEdges) ? eattr[(long)e * HB + lane] : 0.f;
        }
    }
    __syncthreads();

    // WMMA fragment coordinates (wave32 layout).
    const int row  = lane & 15;          // A: M,  B: N,  within half-wave
    const int koff = (lane >> 4) << 1;   // K sub-offset 0 or 2
    const int mb   = (lane >> 4) << 3;   // C/D row base 0 or 8

    // Preload all A fragments for this 16-edge tile (K = 80 -> 20 steps).
    const float* zr = &sZ[wave][row * ZSTR];
    v2f aF[20];
    #pragma unroll
    for (int k = 0; k < 20; ++k) {
        aF[k].x = zr[4 * k + koff];
        aF[k].y = zr[4 * k + koff + 1];
    }

    #pragma unroll
    for (int nt = 0; nt < 4; ++nt) {
        v8f c = {0.f, 0.f, 0.f, 0.f, 0.f, 0.f, 0.f, 0.f};
        #pragma unroll
        for (int k = 0; k < 20; ++k) {
            v2f b;
            b.x = sW2[(4 * k + koff)     * HC + nt * 16 + row];
            b.y = sW2[(4 * k + koff + 1) * HC + nt * 16 + row];
            c = __builtin_amdgcn_wmma_f32_16x16x4_f32(
                    false, aF[k], false, b, (short)0, c, false, false);
        }
        const float bias = b2[nt * 16 + row];
        #pragma unroll
        for (int v = 0; v < 8; ++v) {
            const int M = v + mb;
            const int d = sDst[wave][M];
            if (d >= 0) {
                float val = c[v] + bias;
                val = (val > 0.f) ? val : 0.1f * val;     // LeakyReLU(0.1)
                atomicAdd(&agg[(long)d * HC + nt * 16 + row], val);
            }
        }
    }
}

// ---------------------------------------------------------------------------
// Node update: hout[16x64] = concat(h, agg)[16x128] @ W1[128x64] + b1.
// 128 threads = 4 waves, 16 nodes per wave.
// ---------------------------------------------------------------------------
__global__ void gnn_node_kernel(const float* __restrict__ h,
                                const float* __restrict__ agg,
                                const float* __restrict__ w1,   // [128,64]
                                const float* __restrict__ b1,   // [64]
                                float* __restrict__ hout,
                                int nNodes) {
    __shared__ float sW1[K1 * HC];        // 32 KB

    const int tid  = threadIdx.x;
    const int lane = tid & 31;
    const int wave = tid >> 5;

    for (int i = tid; i < K1 * HC; i += blockDim.x) sW1[i] = w1[i];

    const int nodeBase = blockIdx.x * 64 + wave * 16;
    const int row  = lane & 15;
    const int koff = (lane >> 4) << 1;
    const int mb   = (lane >> 4) << 3;

    const int  node = nodeBase + row;
    const long nrd  = (long)((node < nNodes) ? node : 0) * HC;

    // A fragments: K = 128 -> 32 steps; first 64 from h, rest from agg.
    v2f aF[32];
    #pragma unroll
    for (int k = 0; k < 32; ++k) {
        const int kb = 4 * k + koff;
        const float* sp = (kb < HC) ? h : agg;
        const int o = kb & (HC - 1);
        aF[k].x = sp[nrd + o];
        aF[k].y = sp[nrd + o + 1];
    }
    __syncthreads();

    #pragma unroll
    for (int nt = 0; nt < 4; ++nt) {
        v8f c = {0.f, 0.f, 0.f, 0.f, 0.f, 0.f, 0.f, 0.f};
        #pragma unroll
        for (int k = 0; k < 32; ++k) {
            v2f b;
            b.x = sW1[(4 * k + koff)     * HC + nt * 16 + row];
            b.y = sW1[(4 * k + koff + 1) * HC + nt * 16 + row];
            c = __builtin_amdgcn_wmma_f32_16x16x4_f32(
                    false, aF[k], false, b, (short)0, c, false, false);
        }
        const float bias = b1[nt * 16 + row];
        #pragma unroll
        for (int v = 0; v < 8; ++v) {
            const int nn = nodeBase + v + mb;
            if (nn < nNodes)
                hout[(long)nn * HC + nt * 16 + row] = c[v] + bias;
        }
    }
}

// ---------------------------------------------------------------------------
// Launch: proj -> 3x (memset agg, edge, node).  Ping-pong h in workspace,
// final layer writes d_out.  All ops on `stream` (graph-capture safe).
// ---------------------------------------------------------------------------
extern "C" void kernel_launch(void* const* d_in, const int* in_sizes, int n_in,
                              void* d_out, int out_size, void* d_ws, size_t ws_size,
                              hipStream_t stream) {
    const float* x     = (const float*)d_in[0];
    const int*   ei    = (const int*)  d_in[1];
    const float* eattr = (const float*)d_in[2];
    const float* pw    = (const float*)d_in[3];
    const float* pb    = (const float*)d_in[4];
    const float* u2w   = (const float*)d_in[5];
    const float* u2b   = (const float*)d_in[6];
    const float* u1w   = (const float*)d_in[7];
    const float* u1b   = (const float*)d_in[8];

    const int inDim  = 75;
    const int nNodes = in_sizes[0] / inDim;
    const int nEdges = in_sizes[1] / 2;
    const int* src = ei;
    const int* dst = ei + nEdges;

    const size_t NH = (size_t)nNodes * HC;
    float* bufA = (float*)d_ws;
    float* bufB = bufA + NH;
    float* aggB = bufB + NH;

    // h0 = x @ proj_w + proj_b
    {
        const long total = (long)nNodes * HC;
        const int blocks = (int)((total + 255) / 256);
        gnn_proj_kernel<<<blocks, 256, 0, stream>>>(x, pw, pb, bufA, nNodes, inDim);
    }

    const int edgeBlocks = (nEdges + 63) / 64;
    const int nodeBlocks = (nNodes + 63) / 64;

    for (int l = 0; l < 3; ++l) {
        const float* hin  = (l & 1) ? bufB : bufA;
        float*       hout = (l == 2) ? (float*)d_out : ((l & 1) ? bufA : bufB);

        hipMemsetAsync(aggB, 0, NH * sizeof(float), stream);
        gnn_edge_kernel<<<edgeBlocks, 128, 0, stream>>>(
            hin, src, dst, eattr,
            u2w + (size_t)l * KZ * HC, u2b + (size_t)l * HC,
            aggB, nEdges);
        gnn_node_kernel<<<nodeBlocks, 128, 0, stream>>>(
            hin, aggB,
            u1w + (size_t)l * K1 * HC, u1b + (size_t)l * HC,
            hout, nNodes);
    }
}